// Q6SoftAttentionMask_76055280878200
// MI455X (gfx1250) — compile-verified
//
#include <hip/hip_runtime.h>
#include <hip/hip_bf16.h>
#include <math.h>

typedef __attribute__((ext_vector_type(2))) float v2f;
typedef __attribute__((ext_vector_type(8))) float v8f;

#define T_DIM 2048
#define D_DIM 1024
#define H_DIM 16
#define NT 7
#define NTILE (T_DIM / 16)   // 128

// ---------------------------------------------------------------------------
// Kernel 1: hex_pad[b][t][0..5] = tanh(dot(x[b,t,:], q6_w[e,:])); [6],[7]=0
// One wave32 per (b,t) row. 8 waves per block.
// ---------------------------------------------------------------------------
__global__ __launch_bounds__(256) void hex_kernel(const float* __restrict__ x,
                                                  const float* __restrict__ q6w,
                                                  float* __restrict__ hex_pad) {
  const int row  = blockIdx.x * 8 + (threadIdx.x >> 5);  // b*T + t
  const int lane = threadIdx.x & 31;
  const float* xr = x + (size_t)row * D_DIM;

  float acc[6] = {0.f, 0.f, 0.f, 0.f, 0.f, 0.f};
  for (int d = lane; d < D_DIM; d += 32) {
    const float xv = xr[d];
#pragma unroll
    for (int e = 0; e < 6; ++e) acc[e] += xv * q6w[e * D_DIM + d];
  }
#pragma unroll
  for (int e = 0; e < 6; ++e) {
#pragma unroll
    for (int off = 16; off >= 1; off >>= 1)
      acc[e] += __shfl_down(acc[e], off, 32);
  }
  if (lane == 0) {
    float* o = hex_pad + (size_t)row * 8;
#pragma unroll
    for (int e = 0; e < 6; ++e) o[e] = tanhf(acc[e]);
    o[6] = 0.f;
    o[7] = 0.f;
  }
}

// ---------------------------------------------------------------------------
// Kernel 2: per-head 6x6 combined transform, with sigmoid(scale)*2/6 folded in.
// wt[h][d][e] = (scale/6) * sum_t softmax(tw[h,:]/0.5)[t] * transforms[t][d][e]
// ---------------------------------------------------------------------------
__global__ void wt_kernel(const float* __restrict__ tw,
                          const float* __restrict__ transforms,
                          const float* __restrict__ scale_logit,
                          float* __restrict__ wt) {
  const int h = threadIdx.x;
  if (h >= H_DIM) return;
  float wv[NT];
  float mx = -1e30f;
#pragma unroll
  for (int t = 0; t < NT; ++t) {
    wv[t] = tw[h * NT + t] * 2.0f;  // /TEMPERATURE(0.5)
    mx = fmaxf(mx, wv[t]);
  }
  float s = 0.f;
#pragma unroll
  for (int t = 0; t < NT; ++t) {
    wv[t] = __expf(wv[t] - mx);
    s += wv[t];
  }
  const float scale  = 2.0f / (1.0f + __expf(-scale_logit[0]));
  const float factor = scale * (1.0f / 6.0f) / s;
  for (int de = 0; de < 36; ++de) {
    float acc = 0.f;
#pragma unroll
    for (int t = 0; t < NT; ++t) acc += wv[t] * transforms[t * 36 + de];
    wt[h * 36 + de] = acc * factor;
  }
}

// ---------------------------------------------------------------------------
// Kernel 3: hexT_pad[b][h][i][e] = sum_d hex_pad[b][i][d] * wt[h][d][e]
// (e=6,7 zero-padded). One thread per padded element.
// ---------------------------------------------------------------------------
__global__ __launch_bounds__(256) void hexT_kernel(const float* __restrict__ hex_pad,
                                                   const float* __restrict__ wt,
                                                   float* __restrict__ hexT_pad) {
  const int idx = blockIdx.x * blockDim.x + threadIdx.x;  // B*H*T*8
  const int e   = idx & 7;
  const int bhi = idx >> 3;
  const int i   = bhi & (T_DIM - 1);
  const int bh  = bhi >> 11;  // T=2048
  const int h   = bh & (H_DIM - 1);
  const int b   = bh >> 4;
  float v = 0.f;
  if (e < 6) {
    const float* hx = hex_pad + ((size_t)b * T_DIM + i) * 8;
    const float* W  = wt + h * 36;
#pragma unroll
    for (int d = 0; d < 6; ++d) v += hx[d] * W[d * 6 + e];
  }
  hexT_pad[idx] = v;
}

// ---------------------------------------------------------------------------
// Kernel 4: out[bh][i][j] = sum_{k<8} hexT_pad[bh][i][k] * hex_pad[b][j][k]
// via V_WMMA_F32_16X16X4_F32, two K=4 steps (K=6,7 are zero padding).
// grid = (128 i-tiles, 32 bh); 8 waves/block, each wave does 16 j-tiles.
// ---------------------------------------------------------------------------
__global__ __launch_bounds__(256) void mask_gemm(const float* __restrict__ hexT_pad,  // [B*H][T][8]
                                                 const float* __restrict__ hex_pad,   // [B][T][8]
                                                 float* __restrict__ out) {           // [B*H][T][T]
  const int bh    = blockIdx.y;            // 0..31
  const int b     = bh >> 4;               // H = 16
  const int itile = blockIdx.x;            // 0..127
  const int wave  = threadIdx.x >> 5;      // 0..7
  const int lane  = threadIdx.x & 31;

  const int m0   = itile * 16;
  const int mrow = lane & 15;
  const int half = lane >> 4;              // 0 or 1

  const float* A  = hexT_pad + ((size_t)bh * T_DIM + m0) * 8;
  const float* Bm = hex_pad + (size_t)b * T_DIM * 8;

  // A-frag (16x4, f32): lanes 0-15 hold K = k0+{0,1}; lanes 16-31 K = k0+{2,3}
  const v2f a0 = *(const v2f*)(A + mrow * 8 + 2 * half);       // k0 = 0
  const v2f a1 = *(const v2f*)(A + mrow * 8 + 4 + 2 * half);   // k0 = 4

  float* outbase = out + (size_t)bh * T_DIM * T_DIM + (size_t)m0 * T_DIM;

  for (int jt = wave; jt < NTILE; jt += 8) {
    const int n0 = jt * 16;
    const int n  = n0 + mrow;
    // B-frag (4x16, f32): N = lane&15; half-lanes carry K offset by 2
    const v2f b0 = *(const v2f*)(Bm + n * 8 + 2 * half);
    const v2f b1 = *(const v2f*)(Bm + n * 8 + 4 + 2 * half);

    v8f c = {};
    c = __builtin_amdgcn_wmma_f32_16x16x4_f32(false, a0, false, b0,
                                              (short)0, c, false, false);
    c = __builtin_amdgcn_wmma_f32_16x16x4_f32(false, a1, false, b1,
                                              (short)0, c, false, false);

    // C/D layout: VGPR r -> row m0 + r + 8*half, col n0 + (lane&15)
    float* o = outbase + (size_t)(8 * half) * T_DIM + (n0 + mrow);
#pragma unroll
    for (int r = 0; r < 8; ++r) o[(size_t)r * T_DIM] = c[r];
  }
}

// ---------------------------------------------------------------------------
extern "C" void kernel_launch(void* const* d_in, const int* in_sizes, int n_in,
                              void* d_out, int out_size, void* d_ws, size_t ws_size,
                              hipStream_t stream) {
  const float* x           = (const float*)d_in[0];  // (2,2048,1024)
  const float* q6w         = (const float*)d_in[1];  // (6,1024)
  const float* transforms  = (const float*)d_in[2];  // (7,6,6)
  const float* tw          = (const float*)d_in[3];  // (16,7)
  const float* scale_logit = (const float*)d_in[4];  // scalar
  float* out = (float*)d_out;                        // (2,16,2048,2048)

  char* ws = (char*)d_ws;
  float* hex_pad  = (float*)(ws);                     // 2*2048*8*4   = 128 KB
  float* wt       = (float*)(ws + (128 << 10));       // 16*36*4      ≈ 2.3 KB
  float* hexT_pad = (float*)(ws + (256 << 10));       // 2*16*2048*8*4 = 2 MB

  // 1) hex coords (tanh projection), padded K 6->8
  hex_kernel<<<(2 * T_DIM) / 8, 256, 0, stream>>>(x, q6w, hex_pad);

  // 2) per-head combined 6x6 transform with scale/6 folded in
  wt_kernel<<<1, 32, 0, stream>>>(tw, transforms, scale_logit, wt);

  // 3) transformed hex coords per head, padded K 6->8
  hexT_kernel<<<(2 * H_DIM * T_DIM * 8) / 256, 256, 0, stream>>>(hex_pad, wt, hexT_pad);

  // 4) big bandwidth-bound T x T contraction via f32 WMMA
  dim3 grid(NTILE, 2 * H_DIM);
  mask_gemm<<<grid, 256, 0, stream>>>(hexT_pad, hex_pad, out);
}